// YOLOv1_4569845203301
// MI455X (gfx1250) — compile-verified
//
#include <hip/hip_runtime.h>
#include <stdint.h>

// YOLOv1 post-process: decode + stable score sort + greedy NMS, one image per
// 128-thread block (4 wave32s). Input staged to LDS with CDNA5 async
// global->LDS loads (ASYNCcnt-tracked, no VGPR round trip).

#define NUM_CLASSES 20
#define CH          30      // 2 anchors * 5 + 20 classes
#define CELLS       49      // 7*7
#define NBOX        98      // 49*2
#define IN_PER_IMG  1470    // 49*30
#define NMS_T       0.7f
#define SCORE_T     0.05f
#define CELL_PX     64.0f   // 448/7

__device__ __forceinline__ float sigm(float v) {
    return 1.0f / (1.0f + __expf(-v));
}
__device__ __forceinline__ float clamp01(float v) {
    return fminf(fmaxf(v, 0.0f), 1.0f);
}

// CDNA5 async global->LDS copies (tracked with ASYNCcnt).
__device__ __forceinline__ void async_g2l_b128(uint32_t lds_off, const float* g) {
    asm volatile("global_load_async_to_lds_b128 %0, %1, off"
                 :: "v"(lds_off), "v"(g) : "memory");
}
__device__ __forceinline__ void async_g2l_b32(uint32_t lds_off, const float* g) {
    asm volatile("global_load_async_to_lds_b32 %0, %1, off"
                 :: "v"(lds_off), "v"(g) : "memory");
}
__device__ __forceinline__ void wait_async0() {
    asm volatile("s_wait_asynccnt 0" ::: "memory");
}

__global__ __launch_bounds__(128)
void yolo_post_kernel(const float* __restrict__ x, float* __restrict__ out, int batch) {
    __shared__ float s_in[IN_PER_IMG];
    __shared__ float s_key[NBOX];
    __shared__ float s_x1[NBOX], s_y1[NBOX], s_x2[NBOX], s_y2[NBOX];
    __shared__ float s_score[NBOX], s_area[NBOX];
    __shared__ int   s_label[NBOX];
    __shared__ int   s_alive[NBOX];

    const int img = blockIdx.x;
    if (img >= batch) return;           // uniform per block
    const int tid = threadIdx.x;
    const float* src = x + (size_t)img * IN_PER_IMG;

    // ---- Stage 0: async DMA image into LDS (5880 B = 367*16 + 8) ----
    const uint32_t lds_base = (uint32_t)(uintptr_t)s_in;  // low 32 bits = LDS offset
    for (int c = tid; c < 367; c += 128)
        async_g2l_b128(lds_base + (uint32_t)c * 16u, src + c * 4);
    if (tid < 2)
        async_g2l_b32(lds_base + 5872u + (uint32_t)tid * 4u, src + 1468 + tid);
    wait_async0();          // my wave's copies done
    __syncthreads();        // all waves' copies visible

    // ---- Stage 1: decode box tid (registers) + sort key to LDS ----
    float x1 = 0.f, y1 = 0.f, x2 = 0.f, y2 = 0.f, score = 0.f, key = 0.f;
    int   label = 0, valid = 0;
    if (tid < NBOX) {
        const int cell = tid >> 1, a = tid & 1;
        const int row = cell / 7, col = cell - row * 7;
        const float* p = s_in + cell * CH + a * 5;
        const float bx = sigm(p[0]) + (float)col * CELL_PX;
        const float by = sigm(p[1]) + (float)row * CELL_PX;
        const float bw = sigm(p[2]);
        const float bh = sigm(p[3]);
        valid = (p[4] >= 0.0f);                 // sigmoid(t) >= 0.5  <=>  t >= 0
        x1 = clamp01(bx);      y1 = clamp01(by);
        x2 = clamp01(bx + bw); y2 = clamp01(by + bh);

        const float* q = s_in + cell * CH + 10; // class scores shared per cell
        float m = q[0]; int lab = 0;
        #pragma unroll
        for (int c = 1; c < NUM_CLASSES; ++c) {
            const float v = q[c];
            if (v > m) { m = v; lab = c; }      // first-max, argmax semantics
        }
        score = m; label = lab;
        key = valid ? score : -INFINITY;        // invalid pushed to the end
        s_key[tid] = key;
    }
    __syncthreads();

    // ---- Stage 2: stable rank sort (descending key, ties by index) ----
    if (tid < NBOX) {
        int rank = 0;
        #pragma unroll 7
        for (int k = 0; k < NBOX; ++k) {
            const float kk = s_key[k];
            rank += (int)((kk > key) || (kk == key && k < tid));
        }
        s_x1[rank] = x1;  s_y1[rank] = y1;
        s_x2[rank] = x2;  s_y2[rank] = y2;
        s_score[rank] = score;
        s_label[rank] = label;
        s_area[rank]  = (x2 - x1) * (y2 - y1);
        s_alive[rank] = valid;
    }
    __syncthreads();

    // my sorted box in registers
    float mx1 = 0.f, my1 = 0.f, mx2 = 0.f, my2 = 0.f, marea = 0.f;
    if (tid < NBOX) {
        mx1 = s_x1[tid]; my1 = s_y1[tid];
        mx2 = s_x2[tid]; my2 = s_y2[tid];
        marea = s_area[tid];
    }

    // ---- Stage 3: greedy sequential NMS (matches fori_loop semantics) ----
    #pragma unroll 1
    for (int i = 0; i < NBOX - 1; ++i) {
        __syncthreads();    // writes from iterations < i visible
        if (tid < NBOX && tid > i && s_alive[i] && s_alive[tid]) {
            const float lx = fmaxf(s_x1[i], mx1);
            const float ly = fmaxf(s_y1[i], my1);
            const float rx = fminf(s_x2[i], mx2);
            const float ry = fminf(s_y2[i], my2);
            const float w  = fmaxf(rx - lx, 0.0f);
            const float h  = fmaxf(ry - ly, 0.0f);
            const float inter = w * h;
            const float uni   = s_area[i] + marea - inter;
            const float iou   = (uni > 0.0f) ? (inter / uni) : 0.0f;
            if (iou > NMS_T) s_alive[tid] = 0;   // only thread tid writes slot tid
        }
    }
    __syncthreads();

    // ---- Stage 4: padded outputs (boxes | scores | labels | keep), flat ----
    if (tid < NBOX) {
        const int keep = s_alive[tid] && (s_score[tid] >= SCORE_T);
        const size_t nb = (size_t)batch * NBOX;
        float* boxes  = out;
        float* scores = out + nb * 4;
        float* labels = out + nb * 5;
        float* keepo  = out + nb * 6;
        const size_t o = (size_t)img * NBOX + tid;
        boxes[o * 4 + 0] = keep ? mx1 : 0.0f;
        boxes[o * 4 + 1] = keep ? my1 : 0.0f;
        boxes[o * 4 + 2] = keep ? mx2 : 0.0f;
        boxes[o * 4 + 3] = keep ? my2 : 0.0f;
        scores[o] = keep ? s_score[tid] : 0.0f;
        labels[o] = keep ? (float)s_label[tid] : 0.0f;
        keepo[o]  = keep ? 1.0f : 0.0f;
    }
}

extern "C" void kernel_launch(void* const* d_in, const int* in_sizes, int n_in,
                              void* d_out, int out_size, void* d_ws, size_t ws_size,
                              hipStream_t stream) {
    (void)n_in; (void)out_size; (void)d_ws; (void)ws_size;
    const float* xin = (const float*)d_in[0];
    float* out = (float*)d_out;
    const int batch = in_sizes[0] / IN_PER_IMG;   // 8192
    dim3 grid((unsigned)batch), block(128);
    hipLaunchKernelGGL(yolo_post_kernel, grid, block, 0, stream, xin, out, batch);
}